// NTM_44229573214901
// MI455X (gfx1250) — compile-verified
//
#include <hip/hip_runtime.h>
#include <math.h>
#include <stdint.h>

// NTM forward scan, MI455X (gfx1250, wave32).
// B=64, T=128, O=256, N=65536, M=64, H=512.
#define B_  64
#define T_  128
#define O_  256
#define N_  65536
#define M_  64
#define H_  512
#define EPSF 1e-8f
#define C_W 198   // 3M+6
#define C_R 70    // M+6
#define KCH 2048  // K-chunk per block in read_vec
#define STK 68    // padded LDS row stride (floats): 272B = 16B-aligned, bank stride 4

typedef __attribute__((ext_vector_type(2))) float v2f;
typedef __attribute__((ext_vector_type(8))) float v8f;

__device__ __forceinline__ v8f wmma4(v2f a, v2f b, v8f c) {
  // D(16x16,f32) = A(16x4,f32) * B(4x16,f32) + C
  return __builtin_amdgcn_wmma_f32_16x16x4_f32(
      false, a, false, b, (short)0, c, false, false);
}

// Async global->LDS copy of 16 bytes (ASYNCcnt-tracked, §08 async memory).
__device__ __forceinline__ void async_copy16(void* lds, const float* gsrc) {
  unsigned       loff = (unsigned)(uintptr_t)lds;                  // addr[31:0] = LDS offset
  unsigned long long ga = (unsigned long long)(uintptr_t)gsrc;     // GV mode, saddr=off
  asm volatile("global_load_async_to_lds_b128 %0, %1, off"
               :: "v"(loff), "v"(ga) : "memory");
}
__device__ __forceinline__ void async_wait0() {
  asm volatile("s_wait_asynccnt 0x0" ::: "memory");
}

__device__ __forceinline__ float softplusf(float x) {
  return (x > 20.f) ? x : logf(1.f + expf(x));
}
__device__ __forceinline__ float sigmoidf_(float x) {
  return 1.f / (1.f + expf(-x));
}

// ---------------------------------------------------------------------------
// Head projection: wh = h[b]@W + bias; derive folded params.
//   kb = beta * tanh(k)/(||tanh(k)||+eps); g; s (softmax3); gamma
//   write head: e_s = sigmoid/B, a_s = tanh/B. Optionally zero rt.
// ---------------------------------------------------------------------------
__global__ __launch_bounds__(256) void head_proj(
    const float* __restrict__ h, const float* __restrict__ W,
    const float* __restrict__ bias, int C,
    float* __restrict__ kb, float* __restrict__ gv, float* __restrict__ sv,
    float* __restrict__ gamv, float* __restrict__ ev, float* __restrict__ av,
    float* __restrict__ rt_zero)
{
  __shared__ float hs[H_];
  __shared__ float wh[C_W];
  __shared__ float kk[M_];
  __shared__ float scale[1];
  const int b = blockIdx.x, tid = threadIdx.x;
  for (int i = tid; i < H_; i += 256) hs[i] = h[b * H_ + i];
  __syncthreads();
  for (int j = tid; j < C; j += 256) {
    float acc = bias[j];
    for (int i = 0; i < H_; ++i) acc += hs[i] * W[i * C + j];
    wh[j] = acc;
  }
  __syncthreads();
  if (tid < M_) kk[tid] = tanhf(wh[tid]);
  __syncthreads();
  if (tid == 0) {
    float s = 0.f;
    for (int m = 0; m < M_; ++m) s += kk[m] * kk[m];
    float beta = softplusf(wh[M_]);
    scale[0] = beta / (sqrtf(s) + EPSF);
    gv[b] = sigmoidf_(wh[M_ + 1]);
    float s0 = wh[M_ + 2], s1 = wh[M_ + 3], s2 = wh[M_ + 4];
    float mx = fmaxf(s0, fmaxf(s1, s2));
    float e0 = expf(s0 - mx), e1 = expf(s1 - mx), e2 = expf(s2 - mx);
    float den = e0 + e1 + e2;
    sv[b * 3 + 0] = e0 / den; sv[b * 3 + 1] = e1 / den; sv[b * 3 + 2] = e2 / den;
    gamv[b] = 1.f + softplusf(wh[M_ + 5]);
  }
  __syncthreads();
  if (tid < M_) kb[b * M_ + tid] = kk[tid] * scale[0];
  if (ev != nullptr && tid < M_) {
    ev[b * M_ + tid] = sigmoidf_(wh[C_R + tid]) * (1.0f / B_);
    av[b * M_ + tid] = tanhf(wh[C_R + M_ + tid]) * (1.0f / B_);
  }
  if (rt_zero != nullptr && tid < M_) rt_zero[b * M_ + tid] = 0.f;
}

// rinv[n] = 1/(||mem[n,:]||+eps), one wave per row
__global__ __launch_bounds__(256) void row_norm(const float* __restrict__ mem,
                                                float* __restrict__ rinv)
{
  const int row  = blockIdx.x * 8 + (threadIdx.x >> 5);
  const int lane = threadIdx.x & 31;
  float a = mem[row * M_ + lane];
  float b = mem[row * M_ + 32 + lane];
  float s = a * a + b * b;
  for (int off = 16; off; off >>= 1) s += __shfl_xor(s, off, 32);
  if (lane == 0) rinv[row] = 1.f / (sqrtf(s) + EPSF);
}

// ---------------------------------------------------------------------------
// Content logits: out[b][n] = kb[b,:]·(mem[n,:]*rinv[n]).  WMMA f32, K=64.
// Async-staged LDS tiles, padded stride (conflict-free A/B fragment reads).
// ---------------------------------------------------------------------------
__global__ __launch_bounds__(256) void content_logits(
    const float* __restrict__ kb, const float* __restrict__ mem,
    const float* __restrict__ rinv, float* __restrict__ out)
{
  __shared__ __align__(16) float kbs[B_ * STK];
  __shared__ __align__(16) float mns[32 * STK];
  const int tid = threadIdx.x;
  const int nbase = blockIdx.x * 32;
  for (int c = tid; c < 64 * 16; c += 256) {               // kb: 64 rows x 16 chunks
    int r = c >> 4, q = c & 15;
    async_copy16(&kbs[r * STK + q * 4], kb + r * M_ + q * 4);
  }
  for (int c = tid; c < 32 * 16; c += 256) {               // mem rows (raw)
    int r = c >> 4, q = c & 15;
    async_copy16(&mns[r * STK + q * 4], mem + (size_t)(nbase + r) * M_ + q * 4);
  }
  async_wait0();
  __syncthreads();
  const int wave = tid >> 5, lane = tid & 31;
  const int btile = wave & 3, ntile = wave >> 2;
  const int g = lane >> 4, li = lane & 15;                 // upper lanes: K+2 half
  const float rs = rinv[nbase + ntile * 16 + li];          // per-lane row scale
  v8f acc = {};
  for (int kk = 0; kk < 16; ++kk) {
    const int k0 = kk * 4 + 2 * g;
    v2f a, bf;
    a.x  = kbs[(btile * 16 + li) * STK + k0];
    a.y  = kbs[(btile * 16 + li) * STK + k0 + 1];
    bf.x = mns[(ntile * 16 + li) * STK + k0] * rs;
    bf.y = mns[(ntile * 16 + li) * STK + k0 + 1] * rs;
    acc = wmma4(a, bf, acc);
  }
  const int ncol = nbase + ntile * 16 + li;
  for (int r = 0; r < 8; ++r)
    out[(btile * 16 + r + 8 * g) * N_ + ncol] = acc[r];
}

// ---------------------------------------------------------------------------
// Online softmax partials: 8 chunks per batch row (512 blocks), then combine.
// ---------------------------------------------------------------------------
#define SCHK 8192
__global__ __launch_bounds__(256) void soft_partial(const float* __restrict__ logit,
    float* __restrict__ pmax, float* __restrict__ psum)
{
  __shared__ float sh[256];
  const int b = blockIdx.x >> 3, c = blockIdx.x & 7;
  const float* row = logit + (size_t)b * N_ + c * SCHK;
  const int tid = threadIdx.x;
  float m = -3.4e38f;
  for (int i = tid; i < SCHK; i += 256) m = fmaxf(m, row[i]);
  sh[tid] = m; __syncthreads();
  for (int off = 128; off; off >>= 1) {
    if (tid < off) sh[tid] = fmaxf(sh[tid], sh[tid + off]);
    __syncthreads();
  }
  const float mx = sh[0]; __syncthreads();
  float s = 0.f;
  for (int i = tid; i < SCHK; i += 256) s += expf(row[i] - mx);
  sh[tid] = s; __syncthreads();
  for (int off = 128; off; off >>= 1) {
    if (tid < off) sh[tid] += sh[tid + off];
    __syncthreads();
  }
  if (tid == 0) { pmax[blockIdx.x] = mx; psum[blockIdx.x] = sh[0]; }
}

__global__ __launch_bounds__(64) void soft_combine(const float* __restrict__ pmax,
    const float* __restrict__ psum, float* __restrict__ redMax,
    float* __restrict__ redSum, float* __restrict__ redP)
{
  const int b = threadIdx.x;  // 64 threads
  float mx = -3.4e38f;
  for (int c = 0; c < 8; ++c) mx = fmaxf(mx, pmax[b * 8 + c]);
  float s = 0.f;
  for (int c = 0; c < 8; ++c) s += psum[b * 8 + c] * expf(pmax[b * 8 + c] - mx);
  redMax[b] = mx; redSum[b] = s; redP[b] = 0.f;
}

// ---------------------------------------------------------------------------
// Fused interpolate + shift + sharpen:
//   wg = g*softmax(logit) + (1-g)*w_prev   (in LDS with circular halo)
//   wp = (s0*wg[n-1]+s1*wg[n]+s2*wg[n+1] + eps)^gamma  -> outw (UNnormalized)
//   redP[b] += sum(wp)   (normalizer folded into consumers)
// ---------------------------------------------------------------------------
__global__ __launch_bounds__(256) void interp_sharpen(const float* __restrict__ logit,
    const float* __restrict__ redMax, const float* __restrict__ redSum,
    const float* __restrict__ gv, const float* __restrict__ wprev,
    const float* __restrict__ sv, const float* __restrict__ gamv,
    float* __restrict__ outw, float* __restrict__ redP)
{
  __shared__ float wgs[258];
  __shared__ float sh[256];
  const int b = blockIdx.x >> 8;            // 256 blocks per batch row
  const int n0 = (blockIdx.x & 255) * 256;
  const int tid = threadIdx.x;
  const float mx = redMax[b], sum = redSum[b], g = gv[b];
  const float* lrow = logit + (size_t)b * N_;
  const float* prow = wprev + (size_t)b * N_;
  const int n = n0 + tid;
  wgs[tid + 1] = g * expf(lrow[n] - mx) / sum + (1.f - g) * prow[n];
  if (tid == 0) {
    const int nm = (n0 == 0) ? (N_ - 1) : (n0 - 1);
    wgs[0] = g * expf(lrow[nm] - mx) / sum + (1.f - g) * prow[nm];
  } else if (tid == 1) {
    const int np = (n0 + 256 == N_) ? 0 : (n0 + 256);
    wgs[257] = g * expf(lrow[np] - mx) / sum + (1.f - g) * prow[np];
  }
  __syncthreads();
  const float wt = sv[b * 3 + 0] * wgs[tid] + sv[b * 3 + 1] * wgs[tid + 1]
                 + sv[b * 3 + 2] * wgs[tid + 2];
  const float wp = powf(wt + EPSF, gamv[b]);
  outw[(size_t)b * N_ + n] = wp;
  sh[tid] = wp; __syncthreads();
  for (int off = 128; off; off >>= 1) {
    if (tid < off) sh[tid] += sh[tid + off];
    __syncthreads();
  }
  if (tid == 0) atomicAdd(&redP[b], sh[0]);
}

// fold sharpen normalizer 1/P[b] into e_s/a_s (4096 elements each)
__global__ __launch_bounds__(256) void scale_ea(float* __restrict__ ev,
    float* __restrict__ av, const float* __restrict__ redP)
{
  const int idx = blockIdx.x * 256 + threadIdx.x;   // 16 blocks
  const float inv = 1.f / redP[idx >> 6];
  ev[idx] *= inv; av[idx] *= inv;
}

// ---------------------------------------------------------------------------
// Erase/add via WMMA (K=B=64, 1/B and 1/P prefolded into e/a):
//   mem[n][m] = mem*(1 - w^T@e) + w^T@a, in place.
// ---------------------------------------------------------------------------
__global__ __launch_bounds__(256) void erase_add(const float* __restrict__ w,
    const float* __restrict__ ev, const float* __restrict__ av,
    float* __restrict__ mem)
{
  __shared__ __align__(16) float ws[B_ * 32];   // [b][n-local]
  __shared__ __align__(16) float es[B_ * M_];
  __shared__ __align__(16) float as_[B_ * M_];
  const int tid = threadIdx.x;
  const int nbase = blockIdx.x * 32;
  for (int c = tid; c < 64 * 8; c += 256) {     // ws: 64 rows x 8 chunks
    int b = c >> 3, q = c & 7;
    async_copy16(&ws[b * 32 + q * 4], w + (size_t)b * N_ + nbase + q * 4);
  }
  for (int c = tid; c < 64 * 16; c += 256) {
    int b = c >> 4, q = c & 15;
    async_copy16(&es[b * M_ + q * 4], ev + b * M_ + q * 4);
    async_copy16(&as_[b * M_ + q * 4], av + b * M_ + q * 4);
  }
  async_wait0();
  __syncthreads();
  const int wave = tid >> 5, lane = tid & 31;
  const int ntile = wave & 1, mtile = wave >> 1;
  const int g = lane >> 4, li = lane & 15;
  v8f accE = {}, accA = {};
  for (int kk = 0; kk < 16; ++kk) {
    const int k0 = kk * 4 + 2 * g;
    v2f a, be, ba;
    a.x  = ws[(k0)     * 32 + ntile * 16 + li];   // A[i][k] = w[b=k][n=i]
    a.y  = ws[(k0 + 1) * 32 + ntile * 16 + li];
    be.x = es[(k0)     * M_ + mtile * 16 + li];
    be.y = es[(k0 + 1) * M_ + mtile * 16 + li];
    ba.x = as_[(k0)     * M_ + mtile * 16 + li];
    ba.y = as_[(k0 + 1) * M_ + mtile * 16 + li];
    accE = wmma4(a, be, accE);
    accA = wmma4(a, ba, accA);
  }
  for (int r = 0; r < 8; ++r) {
    const int n = nbase + ntile * 16 + r + 8 * g;
    const int m = mtile * 16 + li;
    const int id = n * M_ + m;
    mem[id] = mem[id] * (1.f - accE[r]) + accA[r];
  }
}

// ---------------------------------------------------------------------------
// r_t = w_r @ mem (64x64 out, K=65536): K split over 32 blocks, async-staged
// 64-wide K slabs, f32 atomic accumulation. (1/P folded in at the consumer.)
// ---------------------------------------------------------------------------
__global__ __launch_bounds__(256) void read_vec(const float* __restrict__ w,
    const float* __restrict__ mem, float* __restrict__ rt)
{
  __shared__ __align__(16) float ws[B_ * STK];  // [b][k-local], padded
  __shared__ __align__(16) float ms[64 * M_];   // [k-local][m]
  const int tid = threadIdx.x;
  const int wave = tid >> 5, lane = tid & 31;
  const int g = lane >> 4, li = lane & 15;
  const int btile = wave & 3, mt = wave >> 2;   // m-tiles mt and mt+2
  v8f acc0 = {}, acc1 = {};
  const int kbase = blockIdx.x * KCH;
  for (int sub = 0; sub < KCH / 64; ++sub) {
    const int k0b = kbase + sub * 64;
    __syncthreads();
    for (int c = tid; c < 64 * 16; c += 256) {
      int r = c >> 4, q = c & 15;
      async_copy16(&ws[r * STK + q * 4], w + (size_t)r * N_ + k0b + q * 4);
      async_copy16(&ms[r * M_ + q * 4], mem + (size_t)(k0b + r) * M_ + q * 4);
    }
    async_wait0();
    __syncthreads();
    for (int kk = 0; kk < 16; ++kk) {
      const int k0 = kk * 4 + 2 * g;
      v2f a, b0, b1;
      a.x  = ws[(btile * 16 + li) * STK + k0];
      a.y  = ws[(btile * 16 + li) * STK + k0 + 1];
      b0.x = ms[(k0)     * M_ + mt * 16 + li];
      b0.y = ms[(k0 + 1) * M_ + mt * 16 + li];
      b1.x = ms[(k0)     * M_ + (mt + 2) * 16 + li];
      b1.y = ms[(k0 + 1) * M_ + (mt + 2) * 16 + li];
      acc0 = wmma4(a, b0, acc0);
      acc1 = wmma4(a, b1, acc1);
    }
  }
  for (int r = 0; r < 8; ++r) {
    const int brow = btile * 16 + r + 8 * g;
    atomicAdd(&rt[brow * M_ + mt * 16 + li], acc0[r]);
    atomicAdd(&rt[brow * M_ + (mt + 2) * 16 + li], acc1[r]);
  }
}

// ---------------------------------------------------------------------------
// Controller: h = relu(x_t@Wxh + (r_t/P)@Wrh + bh); out = sigmoid(h@Wo+bo).
// ---------------------------------------------------------------------------
__global__ __launch_bounds__(256) void controller(
    const float* __restrict__ x, int t, const float* __restrict__ rt,
    const float* __restrict__ redP, float* __restrict__ h,
    const float* __restrict__ Wxh, const float* __restrict__ Wrh,
    const float* __restrict__ bh, const float* __restrict__ Wo,
    const float* __restrict__ bo, float* __restrict__ out)
{
  __shared__ float xs[O_];
  __shared__ float rs[M_];
  __shared__ float hn[H_];
  const int b = blockIdx.x, tid = threadIdx.x;
  for (int i = tid; i < O_; i += 256) xs[i] = x[((size_t)b * T_ + t) * O_ + i];
  if (tid < M_) rs[tid] = rt[b * M_ + tid] / redP[b];
  __syncthreads();
  for (int j = tid; j < H_; j += 256) {
    float acc = bh[j];
    for (int i = 0; i < O_; ++i) acc += xs[i] * Wxh[i * H_ + j];
    for (int m = 0; m < M_; ++m) acc += rs[m] * Wrh[m * H_ + j];
    acc = fmaxf(acc, 0.f);
    hn[j] = acc;
    h[b * H_ + j] = acc;
  }
  __syncthreads();
  for (int j = tid; j < O_; j += 256) {
    float acc = bo[j];
    for (int i = 0; i < H_; ++i) acc += hn[i] * Wo[i * O_ + j];
    out[((size_t)b * T_ + t) * O_ + j] = sigmoidf_(acc);
  }
}

// ---------------------------------------------------------------------------
extern "C" void kernel_launch(void* const* d_in, const int* in_sizes, int n_in,
                              void* d_out, int out_size, void* d_ws, size_t ws_size,
                              hipStream_t stream)
{
  const float* x    = (const float*)d_in[0];
  const float* mem0 = (const float*)d_in[1];
  const float* wr   = (const float*)d_in[2];   // read-head w_prev (eye)
  const float* ww   = (const float*)d_in[3];   // write-head w_prev (eye)
  const float* h0   = (const float*)d_in[4];
  const float* Wxh  = (const float*)d_in[5];
  const float* Wrh  = (const float*)d_in[6];
  const float* bh   = (const float*)d_in[7];
  const float* Wo   = (const float*)d_in[8];
  const float* bo   = (const float*)d_in[9];
  const float* rW   = (const float*)d_in[10];
  const float* rb   = (const float*)d_in[11];
  const float* wW   = (const float*)d_in[12];
  const float* wb   = (const float*)d_in[13];

  // Workspace carve-up (~51 MB of f32).
  float* p   = (float*)d_ws;
  float* mem = p; p += (size_t)N_ * M_;
  float* wA  = p; p += (size_t)B_ * N_;       // logits
  float* wB  = p; p += (size_t)B_ * N_;       // unnormalized sharpened weights
  float* rinv= p; p += N_;
  float* h   = p; p += B_ * H_;
  float* kbW = p; p += B_ * M_;
  float* gW  = p; p += B_;
  float* sW  = p; p += B_ * 3;
  float* gmW = p; p += B_;
  float* eW  = p; p += B_ * M_;
  float* aW  = p; p += B_ * M_;
  float* kbR = p; p += B_ * M_;
  float* gR  = p; p += B_;
  float* sR  = p; p += B_ * 3;
  float* gmR = p; p += B_;
  float* rMax= p; p += B_;
  float* rSum= p; p += B_;
  float* rP  = p; p += B_;
  float* rt  = p; p += B_ * M_;
  float* pmax= p; p += B_ * 8;
  float* psum= p; p += B_ * 8;

  hipMemcpyAsync(mem, mem0, (size_t)N_ * M_ * sizeof(float),
                 hipMemcpyDeviceToDevice, stream);
  hipMemcpyAsync(h, h0, (size_t)B_ * H_ * sizeof(float),
                 hipMemcpyDeviceToDevice, stream);

  for (int t = 0; t < T_; ++t) {
    // ---- write head ----
    head_proj<<<B_, 256, 0, stream>>>(h, wW, wb, C_W, kbW, gW, sW, gmW,
                                      eW, aW, nullptr);
    row_norm<<<N_ / 8, 256, 0, stream>>>(mem, rinv);
    content_logits<<<N_ / 32, 256, 0, stream>>>(kbW, mem, rinv, wA);
    soft_partial<<<B_ * 8, 256, 0, stream>>>(wA, pmax, psum);
    soft_combine<<<1, 64, 0, stream>>>(pmax, psum, rMax, rSum, rP);
    interp_sharpen<<<B_ * 256, 256, 0, stream>>>(wA, rMax, rSum, gW, ww,
                                                 sW, gmW, wB, rP);
    scale_ea<<<16, 256, 0, stream>>>(eW, aW, rP);
    erase_add<<<N_ / 32, 256, 0, stream>>>(wB, eW, aW, mem);
    // ---- read head (uses updated mem) ----
    head_proj<<<B_, 256, 0, stream>>>(h, rW, rb, C_R, kbR, gR, sR, gmR,
                                      nullptr, nullptr, rt /*zero*/);
    row_norm<<<N_ / 8, 256, 0, stream>>>(mem, rinv);
    content_logits<<<N_ / 32, 256, 0, stream>>>(kbR, mem, rinv, wA);
    soft_partial<<<B_ * 8, 256, 0, stream>>>(wA, pmax, psum);
    soft_combine<<<1, 64, 0, stream>>>(pmax, psum, rMax, rSum, rP);
    interp_sharpen<<<B_ * 256, 256, 0, stream>>>(wA, rMax, rSum, gR, wr,
                                                 sR, gmR, wB, rP);
    read_vec<<<N_ / KCH, 256, 0, stream>>>(wB, mem, rt);
    // ---- controller + output ----
    controller<<<B_, 256, 0, stream>>>(x, t, rt, rP, h, Wxh, Wrh, bh, Wo, bo,
                                       (float*)d_out);
  }
}